// DFTB_Layer_73315091742851
// MI455X (gfx1250) — compile-verified
//
#include <hip/hip_runtime.h>
#include <hip/hip_bf16.h>
#include <stdint.h>

// ---------------- problem constants (match reference) ----------------
#define NMOL      512
#define BSIZE     128
#define NVALS     5000000
#define N_DIRECT  2000000
#define N_ROT     1000000
#define M_ROT     3
#define R_OUT     6
#define N_REP     4000000
#define N_Z       5
#define ROT_LEN   (2 + N_DIRECT + N_ROT * R_OUT)   // 8,000,002
#define MAT       (BSIZE * BSIZE)                  // 16384
#define MATN      (NMOL * MAT)                     // 8,388,608

typedef float v2f __attribute__((ext_vector_type(2)));
typedef float v4f __attribute__((ext_vector_type(4)));
typedef float v8f __attribute__((ext_vector_type(8)));
typedef int   gccv4i __attribute__((vector_size(16)));   // matches builtin param type

// ---------------- gfx1250 async global->LDS ----------------
#ifndef __has_builtin
#define __has_builtin(x) 0
#endif
#if __has_builtin(__builtin_amdgcn_global_load_async_to_lds_b32) && \
    __has_builtin(__builtin_amdgcn_global_load_async_to_lds_b128)
#define HAVE_ASYNC_LDS 1
#else
#define HAVE_ASYNC_LDS 0
#endif
#if __has_builtin(__builtin_amdgcn_s_wait_asynccnt)
#define HAVE_WAIT_ASYNC 1
#else
#define HAVE_WAIT_ASYNC 0
#endif

__device__ __forceinline__ void g2l_b32(const float* g, float* l) {
#if HAVE_ASYNC_LDS
    __builtin_amdgcn_global_load_async_to_lds_b32(
        (__attribute__((address_space(1))) int*)(uintptr_t)g,
        (__attribute__((address_space(3))) int*)(unsigned)(uintptr_t)l,
        0, 0);
#else
    *l = *g;
#endif
}

__device__ __forceinline__ void g2l_b128(const float* g, float* l) {
#if HAVE_ASYNC_LDS
    __builtin_amdgcn_global_load_async_to_lds_b128(
        (__attribute__((address_space(1))) gccv4i*)(uintptr_t)g,
        (__attribute__((address_space(3))) gccv4i*)(unsigned)(uintptr_t)l,
        0, 0);
#else
    *(v4f*)l = *(const v4f*)g;
#endif
}

__device__ __forceinline__ void async_wait_all() {
#if HAVE_ASYNC_LDS
#if HAVE_WAIT_ASYNC
    __builtin_amdgcn_s_wait_asynccnt(0);
#else
    asm volatile("s_wait_asynccnt 0x0" ::: "memory");
#endif
#endif
}

// =====================================================================
// K1a: rot_out head + direct gathers
// =====================================================================
__global__ void k_rot_direct(const float* __restrict__ net_vals,
                             const int* __restrict__ g_direct,
                             float* __restrict__ rot_out) {
    int i = blockIdx.x * blockDim.x + threadIdx.x;
    if (i < 2) { rot_out[i] = (float)i; return; }     // [0, 1]
    int d = i - 2;
    if (d < N_DIRECT) rot_out[2 + d] = net_vals[g_direct[d]];
}

// =====================================================================
// K1b: rotated blocks  rot_out[2+N_DIRECT + i*6 + r] = rot_tensor[i,r,:] . vals[i,:]
// =====================================================================
__global__ void k_rot_mat(const float* __restrict__ net_vals,
                          const int* __restrict__ g_mat,
                          const float* __restrict__ rot_tensor,
                          float* __restrict__ rot_out) {
    int t = blockIdx.x * blockDim.x + threadIdx.x;
    if (t >= N_ROT * R_OUT) return;
    int i = t / R_OUT, r = t - i * R_OUT;
    float v0 = net_vals[g_mat[i * 3 + 0]];
    float v1 = net_vals[g_mat[i * 3 + 1]];
    float v2 = net_vals[g_mat[i * 3 + 2]];
    const float* R = rot_tensor + (size_t)(i * R_OUT + r) * 3;
    rot_out[2 + N_DIRECT + t] = fmaf(R[0], v0, fmaf(R[1], v1, R[2] * v2));
}

// =====================================================================
// K2: ep = G @ dQ   (one thread per (mol,row))
// =====================================================================
__global__ void k_ep(const float* __restrict__ G, const float* __restrict__ dQ,
                     float* __restrict__ ep) {
    int t = blockIdx.x * blockDim.x + threadIdx.x;
    if (t >= NMOL * BSIZE) return;
    int m = t >> 7, i = t & 127;
    const float* g = G + (size_t)m * MAT + (size_t)i * BSIZE;
    const float* q = dQ + (size_t)m * BSIZE;
    float s = 0.f;
    #pragma unroll 4
    for (int j = 0; j < BSIZE; ++j) s = fmaf(g[j], q[j], s);
    ep[t] = s;
}

// =====================================================================
// K3: H gather + Fock assembly.  F -> d_out, H -> ws
// =====================================================================
__global__ void k_fock(const float* __restrict__ rot_out,
                       const int* __restrict__ gather_oper,
                       const float* __restrict__ S,
                       const float* __restrict__ ep,
                       float* __restrict__ H, float* __restrict__ F) {
    size_t e = (size_t)blockIdx.x * blockDim.x + threadIdx.x;
    if (e >= (size_t)MATN) return;
    int m = (int)(e >> 14);
    int w = (int)(e & 16383);
    int i = w >> 7, j = w & 127;
    float h = rot_out[gather_oper[e]];
    float epv = ep[m * BSIZE + i] + ep[m * BSIZE + j];
    H[e] = h;
    F[e] = fmaf(-0.5f * S[e], epv, h);
}

// =====================================================================
// K4: Erep segment-sum with LDS-privatized bins
// =====================================================================
__global__ void k_zero(float* __restrict__ p, int n) {
    int i = blockIdx.x * blockDim.x + threadIdx.x;
    if (i < n) p[i] = 0.f;
}

__global__ void k_erep(const float* __restrict__ net_vals,
                       const int* __restrict__ g_rep,
                       const int* __restrict__ seg,
                       float* __restrict__ Erep) {
    __shared__ float bins[NMOL];
    for (int i = threadIdx.x; i < NMOL; i += blockDim.x) bins[i] = 0.f;
    __syncthreads();
    for (int i = blockIdx.x * blockDim.x + threadIdx.x; i < N_REP;
         i += gridDim.x * blockDim.x)
        atomicAdd(&bins[seg[i]], net_vals[g_rep[i]]);
    __syncthreads();
    for (int i = threadIdx.x; i < NMOL; i += blockDim.x)
        atomicAdd(&Erep[i], bins[i]);
}

// =====================================================================
// K5: batched symmetric Jacobi eigensolver, 128x128, one block / molecule.
// =====================================================================
#define JST 129
#define NSWEEP 10

__device__ __forceinline__ void jac_pair(int rnd, int k, int* pp, int* qq) {
    int a, b;
    if (k == 0) { a = rnd % 127; b = 127; }
    else {
        a = (rnd + k) % 127;
        b = (rnd - k + 127 * 4) % 127;
    }
    *pp = a < b ? a : b;
    *qq = a < b ? b : a;
}

__global__ void k_jacobi(const float* __restrict__ Ain,
                         float* __restrict__ wOut,
                         float* __restrict__ Vout) {
    extern __shared__ float smem[];
    float* sA = smem;                 // 128*129
    float* sV = sA + BSIZE * JST;     // 128*129
    float* sc = sV + BSIZE * JST;     // 64
    float* ss = sc + 64;              // 64
    float* sw = ss + 64;              // 128
    __shared__ int srank[BSIZE];

    const int m   = blockIdx.x;
    const int tid = threadIdx.x;
    const float* Am = Ain + (size_t)m * MAT;

    for (int e = tid; e < MAT; e += 256) {
        int r = e >> 7, c = e & 127;
        sA[r * JST + c] = Am[e];
        sV[r * JST + c] = (r == c) ? 1.f : 0.f;
    }
    __syncthreads();

    for (int sweep = 0; sweep < NSWEEP; ++sweep) {
        for (int rnd = 0; rnd < 127; ++rnd) {
            if (tid < 64) {
                int p, q; jac_pair(rnd, tid, &p, &q);
                float apq = sA[p * JST + q];
                float c = 1.f, s = 0.f;
                if (fabsf(apq) > 1e-12f) {
                    float app = sA[p * JST + p], aqq = sA[q * JST + q];
                    float th  = 0.5f * (aqq - app) / apq;
                    float t   = copysignf(1.f, th) /
                                (fabsf(th) + sqrtf(fmaf(th, th, 1.f)));
                    c = rsqrtf(fmaf(t, t, 1.f));
                    s = t * c;
                }
                sc[tid] = c; ss[tid] = s;
            }
            __syncthreads();
            for (int task = tid; task < 64 * BSIZE; task += 256) {
                int k = task >> 7, j = task & 127;
                int p, q; jac_pair(rnd, k, &p, &q);
                float c = sc[k], s = ss[k];
                float ap = sA[p * JST + j], aq = sA[q * JST + j];
                sA[p * JST + j] = fmaf(c, ap, -s * aq);
                sA[q * JST + j] = fmaf(s, ap,  c * aq);
            }
            __syncthreads();
            for (int task = tid; task < 64 * BSIZE; task += 256) {
                int k = task >> 7, i = task & 127;
                int p, q; jac_pair(rnd, k, &p, &q);
                float c = sc[k], s = ss[k];
                float ap = sA[i * JST + p], aq = sA[i * JST + q];
                sA[i * JST + p] = fmaf(c, ap, -s * aq);
                sA[i * JST + q] = fmaf(s, ap,  c * aq);
                float vp = sV[i * JST + p], vq = sV[i * JST + q];
                sV[i * JST + p] = fmaf(c, vp, -s * vq);
                sV[i * JST + q] = fmaf(s, vp,  c * vq);
            }
            __syncthreads();
        }
    }

    if (tid < BSIZE) sw[tid] = sA[tid * JST + tid];
    __syncthreads();
    if (tid < BSIZE) {
        float wi = sw[tid];
        int r = 0;
        for (int j = 0; j < BSIZE; ++j) {
            float wj = sw[j];
            if (wj < wi || (wj == wi && j < tid)) ++r;
        }
        srank[tid] = r;
    }
    __syncthreads();
    if (tid < BSIZE) wOut[m * BSIZE + srank[tid]] = sw[tid];
    for (int e = tid; e < MAT; e += 256) {
        int row = e >> 7, col = e & 127;
        Vout[(size_t)m * MAT + row * BSIZE + srank[col]] = sV[row * JST + col];
    }
}

// =====================================================================
// K6: phiS = Svecs * Svals^{-1/2} (column scaling)
// =====================================================================
__global__ void k_phis(const float* __restrict__ Svecs,
                       const float* __restrict__ Svals,
                       float* __restrict__ phiS) {
    size_t e = (size_t)blockIdx.x * blockDim.x + threadIdx.x;
    if (e >= (size_t)MATN) return;
    int m = (int)(e >> 14), k = (int)(e & 127);
    phiS[e] = Svecs[e] * rsqrtf(Svals[m * BSIZE + k]);
}

// =====================================================================
// K7: batched 128x128x128 matmul, LDS-staged (async global->LDS), f32 WMMA.
//     D = alpha * op(A) * op(B).  Compile-time TA/TB -> branch-free K-loop.
//     Block = 256 threads = 8 waves handles one (molecule, 16-row M strip).
//     LDS: full B panel 128x128 (stride 132 floats, 16B-aligned rows,
//          conflict-free for both K-major and N-major reads) + op(A) strip
//          16x128 (stride 132). 8 waves each produce one 16x16 tile.
// =====================================================================
#define BST 132          // padded LDS row stride (floats); 132*4=528 = 16*33
#define BMM_LDS ((size_t)(BSIZE * BST + 16 * BST) * sizeof(float))

template <int TA, int TB>
__global__ void k_bmm_wmma(const float* __restrict__ A,
                           const float* __restrict__ B,
                           float* __restrict__ D,
                           float alpha) {
    extern __shared__ float smem[];
    float* sB = smem;               // 128 x BST
    float* sAs = sB + BSIZE * BST;  // 16 x BST   (op(A) strip, row-major in M)

    const int tid  = threadIdx.x;
    const int wave = tid >> 5;
    const int lane = tid & 31;
    const int mol  = blockIdx.x >> 3;
    const int sm   = blockIdx.x & 7;            // M strip index (16 rows)
    const int half = lane >> 4;
    const int l16  = lane & 15;

    const float* Am = A + (size_t)mol * MAT;
    const float* Bm = B + (size_t)mol * MAT;
    float*       Dm = D + (size_t)mol * MAT;

    // ---- stage B panel: 4096 float4 quads, perfectly coalesced ----
    for (int e = tid; e < BSIZE * 32; e += 256) {
        int r = e >> 5, c4 = e & 31;
        g2l_b128(Bm + r * BSIZE + c4 * 4, sB + r * BST + c4 * 4);
    }
    // ---- stage op(A) strip (transpose applied at staging time) ----
    if (TA == 0) {
        for (int e = tid; e < 16 * 32; e += 256) {
            int r = e >> 5, c4 = e & 31;
            g2l_b128(Am + (sm * 16 + r) * BSIZE + c4 * 4, sAs + r * BST + c4 * 4);
        }
    } else {
        for (int e = tid; e < 16 * BSIZE; e += 256) {
            int k = e >> 4, r = e & 15;   // 16 consecutive threads -> 64B global
            g2l_b32(Am + k * BSIZE + sm * 16 + r, sAs + r * BST + k);
        }
    }
    async_wait_all();
    __syncthreads();

    // ---- WMMA: wave `wave` computes tile (sm, tn=wave) ----
    const int tn  = wave;
    const int col = tn * 16 + l16;

    v8f acc = {};
    #pragma unroll
    for (int k0 = 0; k0 < BSIZE; k0 += 4) {
        const int ka = k0 + 2 * half;   // lanes 0-15: K=k0,k0+1 ; 16-31: K=k0+2,k0+3
        v2f a, b;
        a.x = sAs[l16 * BST + ka];
        a.y = sAs[l16 * BST + ka + 1];
        if (TB == 0) {
            b.x = sB[ka * BST + col];
            b.y = sB[(ka + 1) * BST + col];
        } else {
            b.x = sB[col * BST + ka];
            b.y = sB[col * BST + ka + 1];
        }
        acc = __builtin_amdgcn_wmma_f32_16x16x4_f32(
            false, a, false, b, (short)0, acc, false, false);
    }
    #pragma unroll
    for (int v = 0; v < 8; ++v) {
        int orow = sm * 16 + v + 8 * half;
        Dm[orow * BSIZE + tn * 16 + l16] = alpha * acc[v];
    }
}

// =====================================================================
// K8: orb_filled = occ_mask * orb
// =====================================================================
__global__ void k_mask(const float* __restrict__ occ,
                       const float* __restrict__ orb,
                       float* __restrict__ of) {
    size_t e = (size_t)blockIdx.x * blockDim.x + threadIdx.x;
    if (e < (size_t)MATN) of[e] = occ[e] * orb[e];
}

// =====================================================================
// K9: Eelec[m] = sum(rho*H) + 0.5 * dQ^T ep   (one block per molecule)
// =====================================================================
__global__ void k_ener(const float* __restrict__ rho,
                       const float* __restrict__ H,
                       const float* __restrict__ dQ,
                       const float* __restrict__ ep,
                       float* __restrict__ Eelec) {
    __shared__ float r1[256], r2[256];
    const int m = blockIdx.x, tid = threadIdx.x;
    const float* r = rho + (size_t)m * MAT;
    const float* h = H   + (size_t)m * MAT;
    float s1 = 0.f;
    for (int e = tid; e < MAT; e += 256) s1 = fmaf(r[e], h[e], s1);
    float s2 = 0.f;
    if (tid < BSIZE) s2 = dQ[m * BSIZE + tid] * ep[m * BSIZE + tid];
    r1[tid] = s1; r2[tid] = s2;
    __syncthreads();
    for (int off = 128; off > 0; off >>= 1) {
        if (tid < off) { r1[tid] += r1[tid + off]; r2[tid] += r2[tid + off]; }
        __syncthreads();
    }
    if (tid == 0) Eelec[m] = r1[0] + 0.5f * r2[0];
}

// =====================================================================
// K10: Eref = zcounts @ eref_vars
// =====================================================================
__global__ void k_eref(const float* __restrict__ zc,
                       const float* __restrict__ ev,
                       float* __restrict__ Eref) {
    int m = blockIdx.x * blockDim.x + threadIdx.x;
    if (m >= NMOL) return;
    float s = 0.f;
    #pragma unroll
    for (int z = 0; z < N_Z; ++z) s = fmaf(zc[m * N_Z + z], ev[z], s);
    Eref[m] = s;
}

// =====================================================================
// launcher
// =====================================================================
extern "C" void kernel_launch(void* const* d_in, const int* in_sizes, int n_in,
                              void* d_out, int out_size, void* d_ws, size_t ws_size,
                              hipStream_t stream) {
    (void)in_sizes; (void)n_in; (void)out_size; (void)ws_size;
    const float* net_vals  = (const float*)d_in[0];
    const int*   g_direct  = (const int*)  d_in[1];
    const int*   g_mat     = (const int*)  d_in[2];
    const float* rot_tens  = (const float*)d_in[3];
    const int*   g_oper    = (const int*)  d_in[4];
    const float* S         = (const float*)d_in[5];
    const float* G         = (const float*)d_in[6];
    const float* dQ        = (const float*)d_in[7];
    const int*   g_rep     = (const int*)  d_in[8];
    const int*   seg_rep   = (const int*)  d_in[9];
    const float* occ       = (const float*)d_in[10];
    const float* zcounts   = (const float*)d_in[11];
    const float* eref_vars = (const float*)d_in[12];

    float* out   = (float*)d_out;
    float* F     = out;                       // NMOL*128*128
    float* Erep  = out + (size_t)MATN;        // 512
    float* Eorb  = Erep + NMOL;               // 512*128
    float* rho   = Eorb + (size_t)NMOL * BSIZE;
    float* Eelec = rho + (size_t)MATN;        // 512
    float* Eref  = Eelec + NMOL;              // 512

    float* ws      = (float*)d_ws;
    float* ep      = ws;                                  // 65,536
    float* wS      = ep + NMOL * BSIZE;                   // 65,536
    float* rot_out = wS + NMOL * BSIZE;                   // 8,000,064 (padded)
    float* B1      = rot_out + 8000064;                   // H
    float* B2      = B1 + (size_t)MATN;                   // Svecs -> T1 -> C -> orb_filled
    float* B3      = B2 + (size_t)MATN;                   // phiS
    float* B4      = B3 + (size_t)MATN;                   // fockp -> orb

    const int TPB = 256;
    const size_t jac_lds = (size_t)(2 * BSIZE * JST + 64 + 64 + BSIZE) * sizeof(float);

    // ---- rotation assembly ----
    k_rot_direct<<<(2 + N_DIRECT + TPB - 1) / TPB, TPB, 0, stream>>>(net_vals, g_direct, rot_out);
    k_rot_mat<<<(N_ROT * R_OUT + TPB - 1) / TPB, TPB, 0, stream>>>(net_vals, g_mat, rot_tens, rot_out);

    // ---- ep, Fock ----
    k_ep<<<(NMOL * BSIZE + TPB - 1) / TPB, TPB, 0, stream>>>(G, dQ, ep);
    k_fock<<<(MATN + TPB - 1) / TPB, TPB, 0, stream>>>(rot_out, g_oper, S, ep, B1, F);

    // ---- Erep ----
    k_zero<<<2, TPB, 0, stream>>>(Erep, NMOL);
    k_erep<<<1024, TPB, 0, stream>>>(net_vals, g_rep, seg_rep, Erep);

    // ---- eigh(S) -> Svals(wS), Svecs(B2) ; phiS(B3) ----
    k_jacobi<<<NMOL, TPB, jac_lds, stream>>>(S, wS, B2);
    k_phis<<<(MATN + TPB - 1) / TPB, TPB, 0, stream>>>(B2, wS, B3);

    // ---- fockp = phiS^T F phiS  (two WMMA matmuls) ----
    k_bmm_wmma<0,0><<<NMOL * 8, TPB, BMM_LDS, stream>>>(F,  B3, B2, 1.0f);  // T1 = F @ phiS
    k_bmm_wmma<1,0><<<NMOL * 8, TPB, BMM_LDS, stream>>>(B3, B2, B4, 1.0f);  // fockp = phiS^T @ T1

    // ---- eigh(fockp) -> Eorb (sorted), C(B2) ----
    k_jacobi<<<NMOL, TPB, jac_lds, stream>>>(B4, Eorb, B2);

    // ---- orb = phiS @ C ; orb_filled ; rho = 2 * of @ of^T ----
    k_bmm_wmma<0,0><<<NMOL * 8, TPB, BMM_LDS, stream>>>(B3, B2, B4, 1.0f);  // orb
    k_mask<<<(MATN + TPB - 1) / TPB, TPB, 0, stream>>>(occ, B4, B2);        // orb_filled
    k_bmm_wmma<0,1><<<NMOL * 8, TPB, BMM_LDS, stream>>>(B2, B2, rho, 2.0f); // rho

    // ---- energies ----
    k_ener<<<NMOL, TPB, 0, stream>>>(rho, B1, dQ, ep, Eelec);
    k_eref<<<(NMOL + TPB - 1) / TPB, TPB, 0, stream>>>(zcounts, eref_vars, Eref);
}